// MNISTIRNN_71889162600448
// MI455X (gfx1250) — compile-verified
//
#include <hip/hip_runtime.h>
#include <math.h>

// MNIST IRNN on MI455X (gfx1250): B=256, H=1024, T=784, C=10, fp32.
//   h_{t+1} = relu(h_t @ Wh^T + bh + x[:,t]*Wi + bi);  out = log_softmax(h @ Wo^T + bo)
//
// Design:
//  * ONE persistent kernel runs all 784 sequential steps (vs 784 graph launches
//    at ~1-2us each, which would rival total compute time).
//  * Wh is invariant across steps. Each WG owns a 32-column slab of Wh:
//    32 x 1024 x 4B = 128KB of LDS, loaded ONCE, pre-swizzled into the exact
//    per-lane V_WMMA_F32_16X16X4_F32 B-fragment layout (fragment = 32 lanes x 8B
//    contiguous -> conflict-free DS loads). All 784 steps feed B from LDS;
//    only h (1MB/step) streams from L2.
//    128KB/WG => 2 WGs per WGP (256KB <= 320KB) => 4 waves/SIMD of latency hiding
//    (vs 2 with a 256KB slab).
//  * fp32 WMMA (16x16x4): reference is fp32 and the IRNN recurrence (Wh ~ I,
//    784 iterations) is precision-sensitive, so no bf16 downcast.
//  * Grid: 64 WGs = 2 (M blocks of 128 batch rows) x 32 (N slabs of 32 cols).
//    8 waves/WG, each wave owns 16M x 32N = 2 accumulator tiles.
//    Per K-step: 1 global b64 (A) + 2 DS fragment loads (B) + 2 wmma.
//  * Grid-wide sync per step: generation barrier, agent-scope atomics + fence.

#define Bsz 256
#define Hsz 1024
#define Tsz 784
#define Csz 10
#define NWG 64

typedef __attribute__((ext_vector_type(2))) float v2f;
typedef __attribute__((ext_vector_type(8))) float v8f;

__device__ __forceinline__ void grid_sync(unsigned* cnt, unsigned* gen) {
    __syncthreads();
    if (threadIdx.x == 0) {
        __threadfence();  // make this WG's h stores visible device-wide
        unsigned g = __hip_atomic_load(gen, __ATOMIC_RELAXED, __HIP_MEMORY_SCOPE_AGENT);
        unsigned arrived = __hip_atomic_fetch_add(cnt, 1u, __ATOMIC_ACQ_REL,
                                                  __HIP_MEMORY_SCOPE_AGENT);
        if (arrived == NWG - 1) {
            __hip_atomic_store(cnt, 0u, __ATOMIC_RELAXED, __HIP_MEMORY_SCOPE_AGENT);
            __hip_atomic_store(gen, g + 1u, __ATOMIC_RELEASE, __HIP_MEMORY_SCOPE_AGENT);
        } else {
            while (__hip_atomic_load(gen, __ATOMIC_ACQUIRE, __HIP_MEMORY_SCOPE_AGENT) == g) {
                __builtin_amdgcn_s_sleep(2);
            }
        }
    }
    __syncthreads();
}

__global__ __launch_bounds__(256) void irnn_persist(
    const float* __restrict__ x,    // [Bsz][Tsz]
    const float* __restrict__ Wi,   // [Hsz]
    const float* __restrict__ bi,   // [Hsz]
    const float* __restrict__ Wh,   // [Hsz][Hsz] row-major Wh[i][j]
    const float* __restrict__ bh,   // [Hsz]
    float* __restrict__ h0,         // [Bsz][Hsz] ping
    float* __restrict__ h1,         // [Bsz][Hsz] pong
    unsigned* __restrict__ sync)    // [2]: arrive count, generation
{
    extern __shared__ float lds[];  // 32*1024 floats = 128KB: swizzled B fragments

    const int tid  = threadIdx.x;
    const int lane = tid & 31;
    const int wave = tid >> 5;          // 0..7 (all distinct M groups)
    const int wg   = blockIdx.x;        // 0..63
    const int nSlab = wg & 31;          // N slab: 32 hidden cols
    const int mBlk  = wg >> 5;          // 0..1 : 128 batch rows
    const int mBase   = mBlk * 128 + wave * 16;
    const int nBaseWG = nSlab * 32;

    // ---------------- Fill B slab, pre-swizzled into fragment layout ----------
    // Fragment fi = ks*2 + nt (ks = k/4 in 0..255, nt = WG-local 16-col tile 0..1).
    // Slot l holds Wh[nBaseWG + nt*16 + (l&15)][ks*4 + ((l>>4)<<1) + e], e=0,1
    // -> per-lane 8B contiguous DS load at (fi*32 + lane)*8 bytes, conflict-free.
    for (int idx = tid; idx < 32 * 1024 / 2; idx += 256) {
        const int l    = idx & 31;
        const int fi   = idx >> 5;      // 0..511
        const int nt   = fi & 1;
        const int ks   = fi >> 1;
        const int col  = nBaseWG + nt * 16 + (l & 15);
        const int k    = ks * 4 + ((l >> 4) << 1);
        const float* src = Wh + (size_t)col * Hsz + k;
        lds[idx * 2 + 0] = src[0];
        lds[idx * 2 + 1] = src[1];
    }

    // Zero this WG's region of h0 (d_ws is poisoned; h0 must start as zeros).
    for (int i = tid; i < 128 * 32; i += 256) {
        const int r = i >> 5, c = i & 31;
        h0[(size_t)(mBlk * 128 + r) * Hsz + nBaseWG + c] = 0.0f;
    }
    grid_sync(&sync[0], &sync[1]);  // all zeros + slabs visible

    // Per-lane constants for A fragment and epilogue.
    const int m    = lane & 15;
    const int koff = (lane >> 4) << 1;
    const int col0 = nBaseWG + m;        // tile 0 output column
    const int col1 = nBaseWG + 16 + m;   // tile 1 output column
    const float wi0 = Wi[col0], c0 = bi[col0] + bh[col0];
    const float wi1 = Wi[col1], c1 = bi[col1] + bh[col1];
    const int rowHalf = (lane >> 4) * 8;

    for (int t = 0; t < Tsz; ++t) {
        const float* hin  = (t & 1) ? h1 : h0;
        float*       hout = (t & 1) ? h0 : h1;

        const float* aPtr = hin + (size_t)(mBase + m) * Hsz + koff;
        v8f acc0 = {}, acc1 = {};
        #pragma unroll 4
        for (int ks = 0; ks < Hsz / 4; ++ks) {
            v2f a  = *(const v2f*)(aPtr + ks * 4);
            v2f b0 = *(const v2f*)(lds + (((ks * 2 + 0) * 32 + lane) << 1));
            v2f b1 = *(const v2f*)(lds + (((ks * 2 + 1) * 32 + lane) << 1));
            acc0 = __builtin_amdgcn_wmma_f32_16x16x4_f32(false, a, false, b0,
                                                         (short)0, acc0, false, false);
            acc1 = __builtin_amdgcn_wmma_f32_16x16x4_f32(false, a, false, b1,
                                                         (short)0, acc1, false, false);
        }

        // Epilogue: D layout -> VGPR r = row (mBase + rowHalf + r), col = lane&15.
        #pragma unroll
        for (int r = 0; r < 8; ++r) {
            const int row = mBase + rowHalf + r;
            const float xv = x[(size_t)row * Tsz + t];
            float v0 = acc0[r] + xv * wi0 + c0;
            float v1 = acc1[r] + xv * wi1 + c1;
            hout[(size_t)row * Hsz + col0] = v0 > 0.0f ? v0 : 0.0f;
            hout[(size_t)row * Hsz + col1] = v1 > 0.0f ? v1 : 0.0f;
        }
        grid_sync(&sync[0], &sync[1]);
    }
}

// One wave per batch row: logits = h[b] @ Wo^T + bo, then log_softmax.
__global__ __launch_bounds__(256) void irnn_head(
    const float* __restrict__ h,    // [Bsz][Hsz]
    const float* __restrict__ Wo,   // [Csz][Hsz]
    const float* __restrict__ bo,   // [Csz]
    float* __restrict__ out)        // [Bsz][Csz]
{
    const int lane = threadIdx.x & 31;
    const int wave = threadIdx.x >> 5;
    const int b = blockIdx.x * 8 + wave;

    float acc[Csz];
    #pragma unroll
    for (int c = 0; c < Csz; ++c) acc[c] = 0.0f;

    for (int j = lane; j < Hsz; j += 32) {
        const float hv = h[(size_t)b * Hsz + j];
        #pragma unroll
        for (int c = 0; c < Csz; ++c) acc[c] += hv * Wo[c * Hsz + j];
    }
    #pragma unroll
    for (int c = 0; c < Csz; ++c)
        for (int off = 16; off > 0; off >>= 1)
            acc[c] += __shfl_down(acc[c], off, 32);

    if (lane == 0) {
        float logits[Csz], mx = -INFINITY;
        #pragma unroll
        for (int c = 0; c < Csz; ++c) {
            logits[c] = acc[c] + bo[c];
            mx = fmaxf(mx, logits[c]);
        }
        float s = 0.0f;
        #pragma unroll
        for (int c = 0; c < Csz; ++c) s += __expf(logits[c] - mx);
        const float lse = mx + __logf(s);
        #pragma unroll
        for (int c = 0; c < Csz; ++c) out[(size_t)b * Csz + c] = logits[c] - lse;
    }
}

extern "C" void kernel_launch(void* const* d_in, const int* in_sizes, int n_in,
                              void* d_out, int out_size, void* d_ws, size_t ws_size,
                              hipStream_t stream) {
    const float* x  = (const float*)d_in[0];   // [256,28,28]
    const float* Wi = (const float*)d_in[1];   // [1024,1]
    const float* bi = (const float*)d_in[2];   // [1024]
    const float* Wh = (const float*)d_in[3];   // [1024,1024]
    const float* bh = (const float*)d_in[4];   // [1024]
    const float* Wo = (const float*)d_in[5];   // [10,1024]
    const float* bo = (const float*)d_in[6];   // [10]
    float* out = (float*)d_out;                // [256,10]

    float* h0 = (float*)d_ws;
    float* h1 = h0 + (size_t)Bsz * Hsz;
    unsigned* sync = (unsigned*)(h1 + (size_t)Bsz * Hsz);

    hipMemsetAsync(sync, 0, 2 * sizeof(unsigned), stream);

    // 128KB dynamic LDS per WG -> two WGs co-resident per WGP (256KB <= 320KB).
    irnn_persist<<<dim3(NWG), dim3(256), 32 * 1024 * sizeof(float), stream>>>(
        x, Wi, bi, Wh, bh, h0, h1, sync);

    // T=784: last step (t=783, odd) writes h0.
    irnn_head<<<dim3(Bsz / 8), dim3(256), 0, stream>>>(h0, Wo, bo, out);
}